// Vmodel_82781199663892
// MI455X (gfx1250) — compile-verified
//
#include <hip/hip_runtime.h>
#include <hip/hip_bf16.h>
#include <math.h>

typedef __attribute__((ext_vector_type(2))) float v2f;
typedef __attribute__((ext_vector_type(4))) float v4f;
typedef __attribute__((ext_vector_type(8))) float v8f;

// ---------------------------------------------------------------------------
// Kernel 1: build the 64x64 v0 table.
//   E[b][c] = (b==c) ? exp(ell[b]) : 1   (elementwise exp of diag(ell))
//   RE_i = R_i @ E ;  tr_term[i,j] = <RE_i, R_j>_frobenius
//   v0[i,j] = exp(-0.5*(trace(E) - tr_term[i,j]))
// ---------------------------------------------------------------------------
__global__ void v0_table_kernel(const float* __restrict__ ell,
                                const float* __restrict__ poses,
                                float* __restrict__ v0) {
    __shared__ float REsh[64][9];
    const int j = threadIdx.x;  // 0..63

    const float e0 = expf(ell[0]);
    const float e1 = expf(ell[1]);
    const float e2 = expf(ell[2]);
    const float eD[3] = {e0, e1, e2};
    const float trE = e0 + e1 + e2;

    float R[9];
#pragma unroll
    for (int a = 0; a < 3; ++a)
#pragma unroll
        for (int b = 0; b < 3; ++b)
            R[a * 3 + b] = poses[j * 12 + a * 4 + b];

#pragma unroll
    for (int a = 0; a < 3; ++a) {
        const float S = R[a * 3 + 0] + R[a * 3 + 1] + R[a * 3 + 2];
#pragma unroll
        for (int c = 0; c < 3; ++c)
            REsh[j][a * 3 + c] = S + R[a * 3 + c] * (eD[c] - 1.0f);
    }
    __syncthreads();

    for (int i = 0; i < 64; ++i) {
        float tr = 0.0f;
#pragma unroll
        for (int k = 0; k < 9; ++k) tr += REsh[i][k] * R[k];
        v0[i * 64 + j] = expf(-0.5f * (trE - tr));
    }
}

// ---------------------------------------------------------------------------
// Kernel 2: 512MB streaming kernel. One wave32 per sample n.
//   xn = x0[d[n]] / ||x0[d[n]]||   (64 floats)
//   wr = v0[w[n]]                   (64 floats)
//   out[n, p, q] = xn[p] * wr[q]    via 16x V_WMMA_F32_16X16X4_F32
//
// Operand roles are TRANSPOSED so stores vectorize:
//   A (16x4, M=q) carries wr chunk in column K=0
//   B (4x16, N=p) carries xn chunk in row    K=0
//   D[M=q, N=p] = wr[q]*xn[p]
// D layout: VGPR r, lanes 0-15 -> M=r, lanes 16-31 -> M=r+8; N = lane%16.
// => lane L's 8 accumulators are 8 CONSECUTIVE q of row p = pi*16+lane%16:
//    two b128 non-temporal stores per tile (32 stores/sample, not 128).
// NT because 512MB output >> 192MB L2.
// ---------------------------------------------------------------------------
#define WAVES_PER_BLOCK 8

__global__ void outer_stream_kernel(const float* __restrict__ x0,
                                    const float* __restrict__ v0,
                                    const int* __restrict__ dIdx,
                                    const int* __restrict__ wIdx,
                                    float* __restrict__ out,
                                    int N) {
    __shared__ float xs[WAVES_PER_BLOCK][64];
    __shared__ float ws[WAVES_PER_BLOCK][64];

    const int lane = threadIdx.x & 31;
    const int wave = threadIdx.x >> 5;
    const int n = blockIdx.x * WAVES_PER_BLOCK + wave;
    const bool active = (n < N);

    if (active) {
        const int di = dIdx[n];
        const int wi = wIdx[n];

        const float xl = x0[(size_t)di * 64 + lane];
        const float xh = x0[(size_t)di * 64 + 32 + lane];

        float ssum = xl * xl + xh * xh;
#pragma unroll
        for (int off = 16; off > 0; off >>= 1)
            ssum += __shfl_xor(ssum, off, 32);
        const float rn = 1.0f / sqrtf(ssum);

        xs[wave][lane]      = xl * rn;
        xs[wave][lane + 32] = xh * rn;
        ws[wave][lane]      = v0[wi * 64 + lane];
        ws[wave][lane + 32] = v0[wi * 64 + 32 + lane];
    }
    __syncthreads();

    if (!active) return;  // wave-uniform: EXEC all-1s for the WMMAs below

    const float* xw = xs[wave];
    const float* ww = ws[wave];
    float* outn = out + (size_t)n * 4096;

    const int half = lane >> 4;   // 0: lanes 0-15, 1: lanes 16-31
    const int l16  = lane & 15;
    // mask-multiply instead of select: unconditional LDS loads, no EXEC masking
    const float msk = (half == 0) ? 1.0f : 0.0f;

    // Build all operand variants ONCE (hoisted out of the tile loops).
    v2f a[4], b[4];
#pragma unroll
    for (int t = 0; t < 4; ++t) {
        a[t].x = ww[t * 16 + l16] * msk;  // A: M dim = q, K=0 column = wr
        a[t].y = 0.0f;                    // K=1 (lanes<16) / K=3 (lanes>=16)
        b[t].x = xw[t * 16 + l16] * msk;  // B: N dim = p, K=0 row = xn
        b[t].y = 0.0f;
    }

#pragma unroll
    for (int pi = 0; pi < 4; ++pi) {
#pragma unroll
        for (int qi = 0; qi < 4; ++qi) {
            v8f c = {};
            // 8 args: (neg_a, A, neg_b, B, c_mod, C, reuse_a, reuse_b)
            v8f dt = __builtin_amdgcn_wmma_f32_16x16x4_f32(
                false, a[qi], false, b[pi], (short)0, c, false, false);

            // element (q = qi*16 + r + 8*half, p = pi*16 + l16)
            // lane-local dt[0..7] = 8 consecutive q -> 2x b128 NT stores
            float* base = outn + (size_t)(pi * 16 + l16) * 64 + qi * 16 + half * 8;
            v4f lo = {dt[0], dt[1], dt[2], dt[3]};
            v4f hi = {dt[4], dt[5], dt[6], dt[7]};
            __builtin_nontemporal_store(lo, (v4f*)base);
            __builtin_nontemporal_store(hi, (v4f*)(base + 4));
        }
    }
}

// ---------------------------------------------------------------------------
// Launch. Inputs: x0 [P*64 f32], ell [3 f32], poses [Q*12 f32],
//                 d [N i32], w [N i32].  Output: [N*4096] f32.
// ---------------------------------------------------------------------------
extern "C" void kernel_launch(void* const* d_in, const int* in_sizes, int n_in,
                              void* d_out, int out_size, void* d_ws, size_t ws_size,
                              hipStream_t stream) {
    const float* x0    = (const float*)d_in[0];
    const float* ell   = (const float*)d_in[1];
    const float* poses = (const float*)d_in[2];
    const int*   d     = (const int*)d_in[3];
    const int*   w     = (const int*)d_in[4];
    float* out = (float*)d_out;
    float* v0  = (float*)d_ws;  // 64*64*4 = 16KB scratch

    v0_table_kernel<<<1, 64, 0, stream>>>(ell, poses, v0);

    const int N = in_sizes[3];  // 32768
    const int blocks = (N + WAVES_PER_BLOCK - 1) / WAVES_PER_BLOCK;
    outer_stream_kernel<<<blocks, 32 * WAVES_PER_BLOCK, 0, stream>>>(
        x0, v0, d, w, out, N);
}